// ExpertGate_85272280695337
// MI455X (gfx1250) — compile-verified
//
#include <hip/hip_runtime.h>

typedef float v2f __attribute__((ext_vector_type(2)));
typedef float v8f __attribute__((ext_vector_type(8)));

#define H_DIM   4096
#define N_EXP   64
#define TOPK    2
#define TOKENS  16384           // 4 * 4096
#define WPB     4               // waves per block
#define TPW     16              // tokens per wave (one WMMA M-tile)
#define LDSS    65              // padded LDS row stride (bank-conflict free)

// d_out layout (floats), reference tuple order
#define O_W   0                               // top_k_weights [T,2]
#define O_P   (TOKENS * TOPK)                 // routing_probs [T,64]
#define O_L   (O_P + TOKENS * N_EXP)          // load_balance_loss scalar
#define O_I   (O_L + 1)                       // top_k_indices [T,2] (as float)

__global__ void ExpertGate_zero_ws(float* ws) {
    ws[threadIdx.x] = 0.0f;                   // 128 floats: weights[64] | counts[64]
}

__global__ __launch_bounds__(WPB * 32)
void ExpertGate_gate_kernel(const float* __restrict__ hs,
                            const float* __restrict__ gw,
                            float* __restrict__ out,
                            float* __restrict__ ws)
{
    __shared__ float ldsExp[WPB][TPW * LDSS];
    __shared__ float ldsInv[WPB][TPW];

    const int wave = threadIdx.x >> 5;
    const int lane = threadIdx.x & 31;
    const int half = lane >> 4;               // 0: lanes 0-15, 1: lanes 16-31
    const int l15  = lane & 15;
    const int tileBase = (blockIdx.x * WPB + wave) * TPW;

    // A-matrix 16x4 f32 layout: lanes 0-15 hold K={0,1}, lanes 16-31 hold K={2,3}
    const float* aPtr = hs + (size_t)(tileBase + l15) * H_DIM + 2 * half;
    // B-matrix 4x16 f32 (row striped across lanes): lane -> expert (l&15), K pair by half
    const float* bPtr = gw + (size_t)l15 * H_DIM + 2 * half;

    v8f acc0 = {}, acc1 = {}, acc2 = {}, acc3 = {};

    #pragma unroll 4
    for (int k = 0; k < H_DIM; k += 4) {
        v2f a  = *(const v2f*)(aPtr + k);
        v2f b0 = *(const v2f*)(bPtr + k);
        v2f b1 = *(const v2f*)(bPtr + (size_t)16 * H_DIM + k);
        v2f b2 = *(const v2f*)(bPtr + (size_t)32 * H_DIM + k);
        v2f b3 = *(const v2f*)(bPtr + (size_t)48 * H_DIM + k);
        // 8 args: (neg_a, A, neg_b, B, c_mod, C, reuse_a, reuse_b)
        acc0 = __builtin_amdgcn_wmma_f32_16x16x4_f32(false, a, false, b0, (short)0, acc0, false, false);
        acc1 = __builtin_amdgcn_wmma_f32_16x16x4_f32(false, a, false, b1, (short)0, acc1, false, false);
        acc2 = __builtin_amdgcn_wmma_f32_16x16x4_f32(false, a, false, b2, (short)0, acc2, false, false);
        acc3 = __builtin_amdgcn_wmma_f32_16x16x4_f32(false, a, false, b3, (short)0, acc3, false, false);
    }

    // C/D layout: VGPR v, lane -> (M = v + 8*half, N = l15). Spill logits to LDS.
    float* myT = &ldsExp[wave][0];
    #pragma unroll
    for (int v = 0; v < 8; ++v) {
        const int row = (v + 8 * half) * LDSS;
        myT[row +  0 + l15] = acc0[v];
        myT[row + 16 + l15] = acc1[v];
        myT[row + 32 + l15] = acc2[v];
        myT[row + 48 + l15] = acc3[v];
    }
    __syncthreads();

    // Per-token softmax + top-2 (one token per lane, lanes 0-15)
    if (lane < TPW) {
        const int t = lane;
        const size_t gt = (size_t)tileBase + t;
        float* rowp = &myT[t * LDSS];

        float mx = -3.402823466e38f;
        for (int e = 0; e < N_EXP; ++e) mx = fmaxf(mx, rowp[e]);

        float s = 0.0f;
        float v1 = -1.0f, v2 = -1.0f;
        int   i1 = 0, i2 = 0;
        for (int e = 0; e < N_EXP; ++e) {
            float ex = __expf(rowp[e] - mx);
            rowp[e] = ex;                      // keep exp values for phase 2b
            s += ex;
            if (ex > v1)      { v2 = v1; i2 = i1; v1 = ex; i1 = e; }
            else if (ex > v2) { v2 = ex; i2 = e; }
        }
        const float inv = 1.0f / s;
        ldsInv[wave][t] = inv;

        const float p1 = v1 * inv, p2 = v2 * inv;
        const float d  = p1 + p2 + 1e-8f;
        const float w1 = p1 / d, w2 = p2 / d;

        out[O_W + gt * 2 + 0] = w1;
        out[O_W + gt * 2 + 1] = w2;
        out[O_I + gt * 2 + 0] = (float)i1;
        out[O_I + gt * 2 + 1] = (float)i2;

        atomicAdd(ws + i1, w1);
        atomicAdd(ws + i2, w2);
        atomicAdd(ws + N_EXP + i1, 1.0f);
        atomicAdd(ws + N_EXP + i2, 1.0f);
    }
    __syncthreads();

    // Coalesced routing_probs writes: 32 lanes cover 64 experts as float2
    #pragma unroll 4
    for (int m = 0; m < TPW; ++m) {
        const float inv = ldsInv[wave][m];
        const float pa = myT[m * LDSS + 2 * lane + 0] * inv;
        const float pb = myT[m * LDSS + 2 * lane + 1] * inv;
        v2f pv = { pa, pb };
        *(v2f*)(out + O_P + (size_t)(tileBase + m) * N_EXP + 2 * lane) = pv;
    }
}

__global__ void ExpertGate_loss_kernel(const float* __restrict__ ws,
                                       float* __restrict__ out)
{
    __shared__ float red[N_EXP];
    const int e = threadIdx.x;
    red[e] = ws[e] * ws[N_EXP + e];
    __syncthreads();
    for (int s = 32; s > 0; s >>= 1) {
        if (e < s) red[e] += red[e + s];
        __syncthreads();
    }
    if (e == 0) {
        const float expected = (float)(TOKENS * TOPK) / (float)N_EXP;   // 512
        out[O_L] = red[0] / (expected * expected);
    }
}

extern "C" void kernel_launch(void* const* d_in, const int* in_sizes, int n_in,
                              void* d_out, int out_size, void* d_ws, size_t ws_size,
                              hipStream_t stream) {
    const float* hs = (const float*)d_in[0];   // hidden_states [4,4096,4096] f32
    const float* gw = (const float*)d_in[1];   // gate_weight   [64,4096]    f32
    float* out = (float*)d_out;
    float* ws  = (float*)d_ws;

    ExpertGate_zero_ws<<<1, 128, 0, stream>>>(ws);
    ExpertGate_gate_kernel<<<TOKENS / (WPB * TPW), WPB * 32, 0, stream>>>(hs, gw, out, ws);
    ExpertGate_loss_kernel<<<1, N_EXP, 0, stream>>>(ws, out);
}